// SpikeNet_3255585211100
// MI455X (gfx1250) — compile-verified
//
#include <hip/hip_runtime.h>

// ---------------- problem constants (match reference) ----------------
#define T_STEPS 8
#define N_NODES 100000
#define D_IN    128
#define B_SEED  20000
#define S1F     5
#define S2F     2
#define H0      256
#define H1      128
#define C_OUT   64
#define ROWS0   (B_SEED * (1 + S1F))   // 120000 rows for layer-0
#define K0      (2 * D_IN)             // 256  (self | neigh concat)
#define K1      (2 * H0)               // 512  (s0_self | s0_neigh_mean)
#define KP      (T_STEPS * H1)         // 1024 pooled feature length

typedef __attribute__((ext_vector_type(2))) float v2f;
typedef __attribute__((ext_vector_type(8))) float v8f;

// ---------------- helpers ----------------
__global__ void zero_ws(float* __restrict__ p, size_t n) {
    size_t i = (size_t)blockIdx.x * blockDim.x + threadIdx.x;
    size_t stride = (size_t)gridDim.x * blockDim.x;
    for (; i < n; i += stride) p[i] = 0.0f;
}

__global__ void concat2(float* __restrict__ dst, const float* __restrict__ a, int na,
                        const float* __restrict__ b, int nb) {
    int i = blockIdx.x * blockDim.x + threadIdx.x;
    if (i < na) dst[i] = a[i];
    else if (i < na + nb) dst[i] = b[i - na];
}

// Build A0[r, 0:128] = x_t[self(r)], A0[r, 128:256] = mean(x_t[nbrs(r)]).
// x_t slice (51MB) is L2-resident on MI455X (192MB L2) -> gathers hit L2.
__global__ void build_A0(const float* __restrict__ x, const int* __restrict__ nodes,
                         const int* __restrict__ nbr1, const int* __restrict__ nbr2,
                         float* __restrict__ A0, int t) {
    int idx = blockIdx.x * blockDim.x + threadIdx.x;
    if (idx >= ROWS0 * 64) return;
    int r = idx >> 6;
    int q = idx & 63;
    const float* xt = x + (size_t)t * N_NODES * D_IN;
    if (q < 32) {                       // self half: one gathered float4
        int c = q * 4;
        int src = (r < B_SEED) ? nodes[r]
                               : nbr1[(size_t)t * B_SEED * S1F + (r - B_SEED)];
        float4 v = *reinterpret_cast<const float4*>(xt + (size_t)src * D_IN + c);
        *reinterpret_cast<float4*>(A0 + (size_t)r * K0 + c) = v;
    } else {                            // neighbor half: fanout mean
        int c = (q - 32) * 4;
        float sx = 0.f, sy = 0.f, sz = 0.f, sw = 0.f;
        if (r < B_SEED) {
            const int* p = nbr1 + ((size_t)t * B_SEED + r) * S1F;
            #pragma unroll
            for (int j = 0; j < S1F; ++j) {
                float4 v = *reinterpret_cast<const float4*>(xt + (size_t)p[j] * D_IN + c);
                sx += v.x; sy += v.y; sz += v.z; sw += v.w;
            }
            sx *= 0.2f; sy *= 0.2f; sz *= 0.2f; sw *= 0.2f;
        } else {
            int i = r - B_SEED;
            const int* p = nbr2 + ((size_t)t * (B_SEED * S1F) + i) * S2F;
            #pragma unroll
            for (int j = 0; j < S2F; ++j) {
                float4 v = *reinterpret_cast<const float4*>(xt + (size_t)p[j] * D_IN + c);
                sx += v.x; sy += v.y; sz += v.z; sw += v.w;
            }
            sx *= 0.5f; sy *= 0.5f; sz *= 0.5f; sw *= 0.5f;
        }
        float4 o; o.x = sx; o.y = sy; o.z = sz; o.w = sw;
        *reinterpret_cast<float4*>(A0 + (size_t)r * K0 + D_IN + c) = o;
    }
}

// Async memory->LDS copy, 16B per lane. Tracked with ASYNCcnt (in-order).
__device__ __forceinline__ void async_cp_b128(unsigned int lds_off, const void* gaddr) {
    asm volatile("global_load_async_to_lds_b128 %0, %1, off"
                 :: "v"(lds_off), "v"(gaddr) : "memory");
}

// ---------------- fp32 WMMA GEMM (V_WMMA_F32_16X16X4_F32) ----------------
// C[M,N] = A[M,K] * B[K,N] (+ bias, + fused LIF epilogues).
// Block tile 128x128, 8 waves, wave tile 64x32 (4x2 frags, 64 VGPRs of acc).
// Double-buffered LDS staged with GLOBAL_LOAD_ASYNC_TO_LDS_B128: next K-tile
// is issued before computing the current one; s_wait_asynccnt 8 (= next tile's
// in-flight ops) guarantees the current tile landed (async loads are in-order).
// OOB rows/cols are index-clamped: garbage feeds only discarded outputs.
// mode 0: LIF(v0) -> dst=A1: self rows direct, neighbor rows atomic mean (s/5)
// mode 1: LIF(v1) -> dst=feats[row*KP + t*H1 + n]
// mode 2: bias only -> dst[row*C_OUT + n]
__launch_bounds__(256)
__global__ void wmma_gemm_f32(const float* __restrict__ A, const float* __restrict__ Bw,
                              int M, int N, int K, int lda, int ldb,
                              const float* __restrict__ bias0, const float* __restrict__ bias1,
                              float* __restrict__ vstate, float* __restrict__ dst,
                              int mode, int tstep) {
    // pad 36: rows stay 16B-aligned for B128 async writes, and 36*i mod 64 is
    // distinct for i=0..15 -> conflict-free frag reads.
    __shared__ float As[2][128][36];
    __shared__ float Bs[2][32][128];

    const int tid   = threadIdx.x;
    const int lane  = tid & 31;
    const int wv    = tid >> 5;
    const int waveM = wv >> 2;      // 0..1
    const int waveN = wv & 3;       // 0..3
    const int l15   = lane & 15;
    const int lh    = lane >> 4;    // lane half: selects K pair / M half per ISA layout

    const int blkM = blockIdx.y * 128;
    const int blkN = blockIdx.x * 128;

    v8f acc[4][2] = {};

    auto issue_tile = [&](int kt, int b) {
        #pragma unroll
        for (int it = 0; it < 4; ++it) {           // A tile 128x32: 4 x b128/lane
            int m  = (tid >> 3) + it * 32;
            int kq = (tid & 7) * 4;
            int grow = blkM + m;
            if (grow >= M) grow = M - 1;           // clamp: stays in-bounds
            const float* g = A + (size_t)grow * lda + kt + kq;
            async_cp_b128((unsigned int)(uintptr_t)&As[b][m][kq], g);
        }
        #pragma unroll
        for (int it = 0; it < 4; ++it) {           // B tile 32x128: 4 x b128/lane
            int idx  = tid + it * 256;
            int kr   = idx >> 5;
            int nq   = (idx & 31) * 4;
            int ncol = blkN + nq;
            if (ncol >= N) ncol = 0;               // clamp: cols >= N discarded later
            const float* g = Bw + (size_t)(kt + kr) * ldb + ncol;
            async_cp_b128((unsigned int)(uintptr_t)&Bs[b][kr][nq], g);
        }
    };

    issue_tile(0, 0);
    int buf = 0;
    for (int k0 = 0; k0 < K; k0 += 32) {
        if (k0 + 32 < K) {
            issue_tile(k0 + 32, buf ^ 1);          // prefetch next tile into other buffer
            asm volatile("s_wait_asynccnt 0x8" ::: "memory");  // current tile done
        } else {
            asm volatile("s_wait_asynccnt 0x0" ::: "memory");
        }
        __syncthreads();                           // all waves' copies visible

        #pragma unroll
        for (int kk = 0; kk < 32; kk += 4) {
            v2f a[4], b[2];
            // A 16x4 frag: lanes 0-15 hold K={kk,kk+1}, lanes 16-31 K={kk+2,kk+3}
            #pragma unroll
            for (int mf = 0; mf < 4; ++mf) {
                int m  = waveM * 64 + mf * 16 + l15;
                int kb = kk + lh * 2;
                a[mf].x = As[buf][m][kb];
                a[mf].y = As[buf][m][kb + 1];
            }
            // B 4x16 frag: N striped across lanes, K pair per lane-half
            #pragma unroll
            for (int nf = 0; nf < 2; ++nf) {
                int n  = waveN * 32 + nf * 16 + l15;
                int kb = kk + lh * 2;
                b[nf].x = Bs[buf][kb][n];
                b[nf].y = Bs[buf][kb + 1][n];
            }
            #pragma unroll
            for (int mf = 0; mf < 4; ++mf)
                #pragma unroll
                for (int nf = 0; nf < 2; ++nf)
                    acc[mf][nf] = __builtin_amdgcn_wmma_f32_16x16x4_f32(
                        false, a[mf], false, b[nf], (short)0, acc[mf][nf], false, false);
        }
        __syncthreads();                           // buffer free for overwrite
        buf ^= 1;
    }

    // epilogue: C/D layout -> vgpr v holds M = m0 + v + lh*8, N = n0 + l15
    #pragma unroll
    for (int mf = 0; mf < 4; ++mf) {
        #pragma unroll
        for (int nf = 0; nf < 2; ++nf) {
            int m0 = blkM + waveM * 64 + mf * 16 + lh * 8;
            int n  = blkN + waveN * 32 + nf * 16 + l15;
            if (n >= N) continue;
            #pragma unroll
            for (int v = 0; v < 8; ++v) {
                int row = m0 + v;
                if (row >= M) continue;
                float val = acc[mf][nf][v];
                if (mode == 0) {
                    float pre = val + bias0[n] + bias1[n];
                    size_t vi = (size_t)row * H0 + n;
                    float u = 0.5f * (vstate[vi] + pre);   // v + (x - v)/tau, tau=2
                    float s = (u >= 1.0f) ? 1.0f : 0.0f;   // spike at VTH=1
                    vstate[vi] = (s > 0.f) ? 0.f : u;      // hard reset
                    if (row < B_SEED) {
                        dst[(size_t)row * K1 + n] = s;     // self half of A1
                    } else if (s > 0.f) {                  // neighbor mean half of A1
                        int bs = (row - B_SEED) / S1F;
                        atomicAdd(&dst[(size_t)bs * K1 + H0 + n], 0.2f);
                    }
                } else if (mode == 1) {
                    float pre = val + bias0[n] + bias1[n];
                    size_t vi = (size_t)row * H1 + n;
                    float u = 0.5f * (vstate[vi] + pre);
                    float s = (u >= 1.0f) ? 1.0f : 0.0f;
                    vstate[vi] = (s > 0.f) ? 0.f : u;
                    dst[(size_t)row * KP + tstep * H1 + n] = s;  // [B, T*H1] layout
                } else {
                    dst[(size_t)row * C_OUT + n] = val + bias0[n];
                }
            }
        }
    }
}

// ---------------- launch ----------------
extern "C" void kernel_launch(void* const* d_in, const int* in_sizes, int n_in,
                              void* d_out, int out_size, void* d_ws, size_t ws_size,
                              hipStream_t stream) {
    (void)in_sizes; (void)n_in; (void)out_size; (void)ws_size;
    const float* x      = (const float*)d_in[0];
    const float* W_l0   = (const float*)d_in[1];
    const float* b_l0   = (const float*)d_in[2];
    const float* W_r0   = (const float*)d_in[3];
    const float* b_r0   = (const float*)d_in[4];
    const float* W_l1   = (const float*)d_in[5];
    const float* b_l1   = (const float*)d_in[6];
    const float* W_r1   = (const float*)d_in[7];
    const float* b_r1   = (const float*)d_in[8];
    const float* W_pool = (const float*)d_in[9];
    const float* b_pool = (const float*)d_in[10];
    const int*   nodes  = (const int*)d_in[11];
    const int*   nbr1   = (const int*)d_in[12];
    const int*   nbr2   = (const int*)d_in[13];
    float* out = (float*)d_out;

    // workspace layout (floats); total ~94.9M floats = ~380 MB
    float* ws    = (float*)d_ws;
    float* A0    = ws;                                  // ROWS0 * K0
    float* A1    = A0 + (size_t)ROWS0 * K0;             // B_SEED * K1
    float* v0    = A1 + (size_t)B_SEED * K1;            // ROWS0 * H0
    float* v1    = v0 + (size_t)ROWS0 * H0;             // B_SEED * H1 (contiguous w/ v0)
    float* feats = v1 + (size_t)B_SEED * H1;            // B_SEED * KP
    float* Wc0   = feats + (size_t)B_SEED * KP;         // K0 * H0
    float* Wc1   = Wc0 + (size_t)K0 * H0;               // K1 * H1

    // zero membrane states (v0 and v1 are contiguous)
    zero_ws<<<2048, 256, 0, stream>>>(v0, (size_t)ROWS0 * H0 + (size_t)B_SEED * H1);

    // concat weights: Wc0 = [W_l0; W_r0] (256x256), Wc1 = [W_l1; W_r1] (512x128)
    concat2<<<(2 * D_IN * H0 + 255) / 256, 256, 0, stream>>>(Wc0, W_l0, D_IN * H0, W_r0, D_IN * H0);
    concat2<<<(2 * H0 * H1 + 255) / 256, 256, 0, stream>>>(Wc1, W_l1, H0 * H1, W_r1, H0 * H1);

    for (int t = 0; t < T_STEPS; ++t) {
        build_A0<<<(ROWS0 * 64 + 255) / 256, 256, 0, stream>>>(x, nodes, nbr1, nbr2, A0, t);
        zero_ws<<<2048, 256, 0, stream>>>(A1, (size_t)B_SEED * K1);
        // layer 0: [120000 x 256] @ [256 x 256], fused LIF -> A1
        wmma_gemm_f32<<<dim3(2, (ROWS0 + 127) / 128), 256, 0, stream>>>(
            A0, Wc0, ROWS0, H0, K0, K0, H0, b_l0, b_r0, v0, A1, 0, t);
        // layer 1: [20000 x 512] @ [512 x 128], fused LIF -> feats[:, t*128:]
        wmma_gemm_f32<<<dim3(1, (B_SEED + 127) / 128), 256, 0, stream>>>(
            A1, Wc1, B_SEED, H1, K1, K1, H1, b_l1, b_r1, v1, feats, 1, t);
    }
    // pool: [20000 x 1024] @ [1024 x 64] + b_pool -> out
    wmma_gemm_f32<<<dim3(1, (B_SEED + 127) / 128), 256, 0, stream>>>(
        feats, W_pool, B_SEED, C_OUT, KP, KP, C_OUT, b_pool, nullptr, nullptr, out, 2, 0);
}